// EKF_1271310319801
// MI455X (gfx1250) — compile-verified
//
#include <hip/hip_runtime.h>
#include <cstdint>
#include <cmath>

typedef uint32_t u32;
typedef uint64_t u64;
typedef u32 u32x4 __attribute__((ext_vector_type(4)));
typedef u32 u32x8 __attribute__((ext_vector_type(8)));
typedef float v2f __attribute__((ext_vector_type(2)));
typedef float v8f __attribute__((ext_vector_type(8)));

#define NSEQ 1024
#define TY   1024
#define CH   32                 // steps per staged chunk
#define NCH  (TY / CH)          // 32 chunks
#define ROW_DW 99               // 96 data dwords + 3 pad dwords (1 per 32, via TDM pad)
#define BUF_DW (32 * ROW_DW)    // one double-buffer half, 32 lanes

// ---------------------------------------------------------------------------
// Tensor Data Mover: stage a 2D tile (32 rows x 96 fp32) global -> LDS.
//   group0: count=1 | lds_addr | global_addr(57b) | type=2
//   group1: data_size=4B, pad_enable, pad_interval=32dw, pad_amount=1dw,
//           tensor_dim0=96, tensor_dim1=32, tile_dim0=96, tile_dim1=32,
//           tensor_dim0_stride=row_stride
// Tracked with TENSORcnt.
// ---------------------------------------------------------------------------
__device__ __forceinline__ void tdm_load_tile(u64 gaddr, u32 lds_addr, u32 row_stride) {
  u32x4 g0;
  g0[0] = 1u;                                               // count=1 (valid, user mode)
  g0[1] = lds_addr;                                         // LDS byte address
  g0[2] = (u32)gaddr;                                       // global addr [31:0]
  g0[3] = ((u32)(gaddr >> 32) & 0x01FFFFFFu) | (2u << 30);  // addr[56:32] | type=2
  u32x8 g1;
  g1[0] = (2u << 16) | (1u << 20) | (4u << 22);             // 4B elems, pad 1dw every 32dw
  g1[1] = (96u << 16);                                      // tensor_dim0 = 96
  g1[2] = (32u << 16);                                      // tensor_dim1 = 32
  g1[3] = (96u << 16);                                      // tile_dim0 = 96
  g1[4] = 32u;                                              // tile_dim1 = 32, tile_dim2 = 0
  g1[5] = row_stride;                                       // tensor_dim0_stride [31:0]
  g1[6] = 0u;
  g1[7] = 0u;
  u32x4 gz = 0;                                             // groups 2/3: zeroed (2D tile)
  asm volatile("tensor_load_to_lds %0, %1, %2, %3"
               :: "s"(g0), "s"(g1), "s"(gz), "s"(gz)
               : "memory");
}

// LDS index of element k (0..95) within a padded 99-dword row.
__device__ __forceinline__ int pad_idx(int k) { return k + (k >> 5); }

__global__ __launch_bounds__(32) void ekf_main(const float* __restrict__ X,
                                               const float* __restrict__ Y,
                                               const float* __restrict__ Qm,
                                               const float* __restrict__ Rm,
                                               float* __restrict__ out,
                                               float* __restrict__ ws) {
  __shared__ __align__(16) float sY[2 * BUF_DW];
  __shared__ __align__(16) float sX[2 * BUF_DW];

  const int lane = threadIdx.x;
  const int blk  = blockIdx.x;
  const int seq  = blk * 32 + lane;

  // Uniform (per-block) TDM bases: 32 consecutive sequences form the tile rows.
  const u64 ybase = (u64)(uintptr_t)Y + (u64)blk * (32ull * TY * 3ull * 4ull);
  const u64 xbase = (u64)(uintptr_t)X + ((u64)blk * (32ull * (TY + 1) * 3ull) + 3ull) * 4ull; // t=1
  const u32 ldsY  = (u32)(uintptr_t)&sY[0];
  const u32 ldsX  = (u32)(uintptr_t)&sX[0];

  // Prologue: stage chunks 0 and 1 (double buffered). 2 tensor ops per chunk.
  tdm_load_tile(ybase,            ldsY,               (u32)(TY * 3));
  tdm_load_tile(xbase,            ldsX,               (u32)((TY + 1) * 3));
  tdm_load_tile(ybase + CH * 12u, ldsY + BUF_DW * 4u, (u32)(TY * 3));
  tdm_load_tile(xbase + CH * 12u, ldsX + BUF_DW * 4u, (u32)((TY + 1) * 3));

  // Lorenz Jacobian constants: F = I + DT*J
  constexpr float DT = 0.02f, SG = 10.0f, RH = 28.0f, BT = 8.0f / 3.0f;
  constexpr float Fa = 1.0f - DT * SG;   // F00
  constexpr float Fb = DT * SG;          // F01
  constexpr float Fd = 1.0f - DT;        // F11
  constexpr float Fh = 1.0f - DT * BT;   // F22

  const float q0 = Qm[0], q1 = Qm[4], q2v = Qm[8];   // Q diag
  const float r0 = Rm[0], r1 = Rm[4], r2v = Rm[8];   // R diag

  float* traj = out;
  float* Pk   = out + (size_t)NSEQ * (TY + 1) * 3;

  // t = 0 rows are zero in the reference output.
  {
    const size_t tb = (size_t)seq * (TY + 1);
    float* tp = traj + tb * 3;
    tp[0] = 0.f; tp[1] = 0.f; tp[2] = 0.f;
    float* pp = Pk + tb * 9;
#pragma unroll
    for (int k = 0; k < 9; ++k) pp[k] = 0.f;
  }

  // Filter state (registers only): x, symmetric P as row pairs + scalars.
  float x0 = 1.f, x1 = 1.f, x2 = 1.f;
  float p00 = 1.f, p01 = 0.f, p02 = 0.f, p11 = 1.f, p12 = 0.f, p22 = 1.f;
  float acc = 0.f;

  for (int c = 0; c < NCH; ++c) {
    // Wait for this chunk's 2 tensor ops (keep the newer chunk's 2 in flight).
    if (c == NCH - 1) { asm volatile("s_wait_tensorcnt 0x0" ::: "memory"); }
    else              { asm volatile("s_wait_tensorcnt 0x2" ::: "memory"); }

    const int b = c & 1;
    const float* ybuf = &sY[b * BUF_DW + lane * ROW_DW];
    const float* xbuf = &sX[b * BUF_DW + lane * ROW_DW];

#pragma unroll 4
    for (int i = 0; i < CH; ++i) {
      const int k0 = 3 * i;
      const float y0 = ybuf[pad_idx(k0)];
      const float y1 = ybuf[pad_idx(k0 + 1)];
      const float y2 = ybuf[pad_idx(k0 + 2)];
      const float gx0 = xbuf[pad_idx(k0)];
      const float gx1 = xbuf[pad_idx(k0 + 1)];
      const float gx2 = xbuf[pad_idx(k0 + 2)];

      // --- predict ---------------------------------------------------------
      const float Fc = DT * (RH - x2);   // F10
      const float Fe = -DT * x0;         // F12
      const float Ff = DT * x1;          // F20
      const float Fg = DT * x0;          // F21

      const float xn0 = x0 + DT * (SG * (x1 - x0));
      const float xn1 = x1 + DT * (x0 * (RH - x2) - x1);
      const float xn2 = x2 + DT * (x0 * x1 - BT * x2);

      // P rows: pair (cols 0,1) + scalar (col 2)  -> v_pk_fma_f32 friendly
      const v2f R0 = {p00, p01}; const float r0s = p02;
      const v2f R1 = {p01, p11}; const float r1s = p12;
      const v2f R2 = {p02, p12}; const float r2s = p22;

      // M = F * P (F02 == 0)
      const v2f M0 = Fa * R0 + Fb * R1;           const float m02 = Fa * r0s + Fb * r1s;
      const v2f M1 = Fc * R0 + Fd * R1 + Fe * R2; const float m12 = Fc * r0s + Fd * r1s + Fe * r2s;
      const v2f M2 = Ff * R0 + Fg * R1 + Fh * R2; const float m22 = Ff * r0s + Fg * r1s + Fh * r2s;

      // G = F^T rows: pair (F[0][k],F[1][k]) + scalar F[2][k]
      const v2f G0 = {Fa, Fc}; const float g0s = Ff;
      const v2f G1 = {Fb, Fd}; const float g1s = Fg;
      const v2f G2 = {0.f, Fe}; const float g2s = Fh;

      // P- = M * F^T + Q   (rows 0,1 as pairs; row 2 scalar)
      const v2f Qv0 = {q0, 0.f};
      const v2f Qv1 = {0.f, q1};
      const v2f PN0 = M0.x * G0 + M0.y * G1 + m02 * G2 + Qv0;
      const float pn02 = M0.x * g0s + M0.y * g1s + m02 * g2s;
      const v2f PN1 = M1.x * G0 + M1.y * G1 + m12 * G2 + Qv1;
      const float pn12 = M1.x * g0s + M1.y * g1s + m12 * g2s;
      const float pn22 = M2.x * g0s + M2.y * g1s + m22 * g2s + q2v;

      const float pn00 = PN0.x, pn01 = PN0.y, pn11 = PN1.y;

      // --- update (H = I): Re = P- + R, K = P- * Re^-1 ---------------------
      const float e00 = pn00 + r0, e01 = pn01, e02 = pn02;
      const float e11 = pn11 + r1, e12 = pn12, e22 = pn22 + r2v;

      const float c00 = e11 * e22 - e12 * e12;
      const float c01 = e02 * e12 - e01 * e22;
      const float c02 = e01 * e12 - e02 * e11;
      const float det = e00 * c00 + e01 * c01 + e02 * c02;
      const float idet = __builtin_amdgcn_rcpf(det);
      const float i00 = c00 * idet, i01 = c01 * idet, i02 = c02 * idet;
      const float i11 = (e00 * e22 - e02 * e02) * idet;
      const float i12 = (e01 * e02 - e00 * e12) * idet;
      const float i22 = (e00 * e11 - e01 * e01) * idet;

      // inv(Re) rows as pairs + scalars
      const v2f I0 = {i00, i01}; const float i0s = i02;
      const v2f I1 = {i01, i11}; const float i1s = i12;
      const v2f I2 = {i02, i12}; const float i2s = i22;

      // K rows = P- rows * inv(Re)
      const v2f K0 = pn00 * I0 + pn01 * I1 + pn02 * I2;
      const float k02 = pn00 * i0s + pn01 * i1s + pn02 * i2s;
      const v2f K1 = pn01 * I0 + pn11 * I1 + pn12 * I2;
      const float k12 = pn01 * i0s + pn11 * i1s + pn12 * i2s;
      const v2f K2 = pn02 * I0 + pn12 * I1 + pn22 * I2;
      const float k22 = pn02 * i0s + pn12 * i1s + pn22 * i2s;

      const float in0 = y0 - xn0, in1 = y1 - xn1, in2 = y2 - xn2;
      x0 = xn0 + K0.x * in0 + K0.y * in1 + k02 * in2;
      x1 = xn1 + K1.x * in0 + K1.y * in1 + k12 * in2;
      x2 = xn2 + K2.x * in0 + K2.y * in1 + k22 * in2;

      // P+ = P- - K * P-   (rows 0,1 as pairs; row 2 scalar; symmetric)
      const v2f PR1 = {pn01, pn11};
      const v2f PR2 = {pn02, pn12};
      const v2f P0n = PN0 - (K0.x * PN0 + K0.y * PR1 + k02 * PR2);
      const float p02n = pn02 - (K0.x * pn02 + K0.y * pn12 + k02 * pn22);
      const v2f P1n = PR1 - (K1.x * PN0 + K1.y * PR1 + k12 * PR2);
      const float p12n = pn12 - (K1.x * pn02 + K1.y * pn12 + k12 * pn22);
      const float p22n = pn22 - (K2.x * pn02 + K2.y * pn12 + k22 * pn22);

      p00 = P0n.x; p01 = P0n.y; p02 = p02n;
      p11 = P1n.y; p12 = p12n;  p22 = p22n;

      // --- outputs ---------------------------------------------------------
      const int t = c * CH + i;
      const size_t ot = (size_t)seq * (TY + 1) + t + 1;
      float* tp = traj + ot * 3;
      tp[0] = x0; tp[1] = x1; tp[2] = x2;
      float* pp = Pk + ot * 9;
      pp[0] = p00; pp[1] = p01; pp[2] = p02;
      pp[3] = p01; pp[4] = p11; pp[5] = p12;
      pp[6] = p02; pp[7] = p12; pp[8] = p22;

      const float d0 = gx0 - x0, d1 = gx1 - x1, d2 = gx2 - x2;
      acc += d0 * d0 + d1 * d1 + d2 * d2;
    }

    // Refill the just-consumed buffer with chunk c+2.
    if (c + 2 < NCH) {
      asm volatile("s_wait_dscnt 0x0" ::: "memory");  // LDS reads done before DMA overwrite
      const u64 off = (u64)(c + 2) * (CH * 12u);
      tdm_load_tile(ybase + off, ldsY + (u32)(b * BUF_DW * 4), (u32)(TY * 3));
      tdm_load_tile(xbase + off, ldsX + (u32)(b * BUF_DW * 4), (u32)((TY + 1) * 3));
    }
  }

  // Per-sequence MSE in dB (mean over Ty*3 = 3072 elements).
  ws[seq] = 10.0f * log10f(acc * (1.0f / 3072.0f));
}

// ---------------------------------------------------------------------------
// Deterministic matrix-engine reduction of the 1024 per-sequence dB values:
// 16x V_WMMA_F32_16X16X4_F32 with B = ones accumulate row-sums of 16x4 tiles.
// D[i][j] = sum_chunks rowsum_i(A) for every column j, so lane0 (rows 0-7,
// col 0) + lane16 (rows 8-15, col 0) hold the full sum. Fixed-order, so the
// result is identical on every replay.
// ---------------------------------------------------------------------------
__global__ __launch_bounds__(32) void ekf_reduce(const float* __restrict__ ws,
                                                 float* __restrict__ outv) {
  const int l = threadIdx.x;
  const int r = l & 15;                 // A-matrix row this lane feeds
  const int koff = (l < 16) ? 0 : 2;    // K columns 0,1 (lanes 0-15) / 2,3 (16-31)

  const v2f bones = {1.0f, 1.0f};       // B = all ones (layout-invariant)
  v8f cacc = {};
#pragma unroll
  for (int m = 0; m < 16; ++m) {
    v2f a;
    a.x = ws[m * 64 + r * 4 + koff];
    a.y = ws[m * 64 + r * 4 + koff + 1];
    cacc = __builtin_amdgcn_wmma_f32_16x16x4_f32(false, a, false, bones,
                                                 (short)0, cacc, false, false);
  }
  float s = cacc[0] + cacc[1] + cacc[2] + cacc[3] +
            cacc[4] + cacc[5] + cacc[6] + cacc[7];

  __shared__ float part[32];
  part[l] = s;
  __syncthreads();
  if (l == 0) outv[0] = (part[0] + part[16]) * (1.0f / 1024.0f);
}

extern "C" void kernel_launch(void* const* d_in, const int* in_sizes, int n_in,
                              void* d_out, int out_size, void* d_ws, size_t ws_size,
                              hipStream_t stream) {
  (void)in_sizes; (void)n_in; (void)out_size; (void)ws_size;
  const float* X  = (const float*)d_in[0];
  const float* Y  = (const float*)d_in[1];
  const float* Qm = (const float*)d_in[2];
  const float* Rm = (const float*)d_in[3];
  float* out = (float*)d_out;
  float* ws  = (float*)d_ws;

  hipLaunchKernelGGL(ekf_main, dim3(NSEQ / 32), dim3(32), 0, stream,
                     X, Y, Qm, Rm, out, ws);
  hipLaunchKernelGGL(ekf_reduce, dim3(1), dim3(32), 0, stream,
                     ws, out + (size_t)NSEQ * (TY + 1) * 12);
}